// BinaryTreeTensorNetwork_31001073942984
// MI455X (gfx1250) — compile-verified
//
#include <hip/hip_runtime.h>
#include <math.h>
#include <stdint.h>

#define TDEPTH  14
#define NLEAVES (1 << TDEPTH)   // 16384
#define TBLOCK  256

#ifndef USE_ASYNC_LDS
#define USE_ASYNC_LDS 1
#endif

struct TreeParams {
    const float* W[TDEPTH];   // layer k weights, flat (n_nodes*2)
    const float* B[TDEPTH];   // layer k biases, flat (n_nodes)
};

__device__ __forceinline__ float fast_tanh(float v) {
#if defined(__AMDGCN__) && __has_builtin(__builtin_amdgcn_tanhf)
    return __builtin_amdgcn_tanhf(v);   // gfx1250 v_tanh_f32 (trans pipe)
#else
    return tanhf(v);
#endif
}

#if USE_ASYNC_LDS
// Async global->LDS copy, 16B per lane. ldsOff = wave-relative LDS byte addr
// (low 32 bits of a generic LDS pointer, per ISA section 10.2), gOff = per-lane
// byte offset, base = uniform 64-bit base address (SGPR pair).
#define ASYNC_LD_B128(ldsOff, gOff, base)                                  \
    asm volatile("global_load_async_to_lds_b128 %0, %1, %2"                \
                 :: "v"(ldsOff), "v"(gOff), "s"(base) : "memory")
#endif

// One workgroup per batch row: 14-layer binary-tree reduction of 16384 leaves.
__global__ __launch_bounds__(TBLOCK) void btt_tree_kernel(
        const float* __restrict__ x, TreeParams p, float* __restrict__ out) {
    const int t   = threadIdx.x;
    const int row = blockIdx.x;
    const float* __restrict__ xr = x + (size_t)row * NLEAVES;

    __shared__ float lvl2[4096 + 256];  // layer-2 nodes, padded: m -> m + (m>>4)
    __shared__ float tailA[256];
    __shared__ float tailB[128];

    const float4* __restrict__ w1 = (const float4*)p.W[0];
    const float2* __restrict__ w2 = (const float2*)p.W[1];
    const float2* __restrict__ b1 = (const float2*)p.B[0];
    const float*  __restrict__ b2 = p.B[1];

    // ---------------- Phase 1: layers 1-2 (global -> registers -> LDS) ------
#if USE_ASYNC_LDS
    __shared__ float stage[2][2048];    // 2 x 8KB double buffer
    const unsigned long long xbase = (unsigned long long)(uintptr_t)xr;
    const unsigned sb0 = (unsigned)(uintptr_t)&stage[0][0];
    const unsigned sb1 = (unsigned)(uintptr_t)&stage[1][0];

    // prime chunk 0 (2048 floats = 2 async b128 issues per lane)
    {
        unsigned l = sb0 + (unsigned)t * 16u;
        unsigned g = (unsigned)t * 16u;
        ASYNC_LD_B128(l,          g,          xbase);
        ASYNC_LD_B128(l + 4096u,  g + 4096u,  xbase);
    }
    for (int c = 0; c < 8; ++c) {
        if (c + 1 < 8) {
            unsigned sb = ((c + 1) & 1) ? sb1 : sb0;
            unsigned l  = sb + (unsigned)t * 16u;
            unsigned g  = (unsigned)(c + 1) * 8192u + (unsigned)t * 16u;
            ASYNC_LD_B128(l,         g,         xbase);
            ASYNC_LD_B128(l + 4096u, g + 4096u, xbase);
            asm volatile("s_wait_asynccnt 2" ::: "memory");  // chunk c landed
        } else {
            asm volatile("s_wait_asynccnt 0" ::: "memory");
        }
        __syncthreads();   // all 8 waves have their chunk-c data in LDS
        const float4* s4 = (const float4*)stage[c & 1];
        #pragma unroll
        for (int q = 0; q < 2; ++q) {
            const int m = c * 512 + q * 256 + t;       // float4 group index
            float4 xv  = s4[q * 256 + t];
            float4 wv  = w1[m];
            float2 w2v = w2[m];
            float2 b1v = b1[m];
            float  b2v = b2[m];
            float a0 = fast_tanh(fmaf(wv.x, xv.x, fmaf(wv.y, xv.y, b1v.x)));
            float a1 = fast_tanh(fmaf(wv.z, xv.z, fmaf(wv.w, xv.w, b1v.y)));
            lvl2[m + (m >> 4)] =
                fast_tanh(fmaf(w2v.x, a0, fmaf(w2v.y, a1, b2v)));
        }
        __syncthreads();   // buffer (c&1) free for reuse at chunk c+2
    }
#else
    const float4* __restrict__ x4 = (const float4*)xr;
    #pragma unroll 4
    for (int i = 0; i < 16; ++i) {
        const int m = i * 256 + t;                     // coalesced float4 group
        float4 xv  = x4[m];
        if (i < 15) __builtin_prefetch(&x4[m + 256], 0, 0);  // global_prefetch_b8
        float4 wv  = w1[m];
        float2 w2v = w2[m];
        float2 b1v = b1[m];
        float  b2v = b2[m];
        float a0 = fast_tanh(fmaf(wv.x, xv.x, fmaf(wv.y, xv.y, b1v.x)));
        float a1 = fast_tanh(fmaf(wv.z, xv.z, fmaf(wv.w, xv.w, b1v.y)));
        lvl2[m + (m >> 4)] = fast_tanh(fmaf(w2v.x, a0, fmaf(w2v.y, a1, b2v)));
    }
    __syncthreads();
#endif

    // ---------------- Phase 2: layers 3-6 (LDS -> registers) ---------------
    float a[16];
    #pragma unroll
    for (int i = 0; i < 16; ++i)
        a[i] = lvl2[17 * t + i];        // element 16t+i at padded index 17t+i

    #pragma unroll
    for (int lay = 2; lay < 6; ++lay) { // W[2]..W[5] = layers 3..6
        const int nout = 1 << (5 - lay);
        const float* __restrict__ W  = p.W[lay];
        const float* __restrict__ Bv = p.B[lay];
        #pragma unroll
        for (int j = 0; j < (1 << (5 - lay)); ++j) {
            const int n = nout * t + j;
            a[j] = fast_tanh(fmaf(W[2 * n], a[2 * j],
                             fmaf(W[2 * n + 1], a[2 * j + 1], Bv[n])));
        }
    }
    tailA[t] = a[0];                    // layer-6 node t (256 nodes)
    __syncthreads();

    // ---------------- Phase 3: layers 7-14 (LDS tree, ping-pong) -----------
    float* src = tailA;
    float* dst = tailB;
    #pragma unroll
    for (int lay = 6; lay < TDEPTH; ++lay) {
        const int nout = 1 << (TDEPTH - 1 - lay);   // 128,64,...,1
        if (t < nout) {
            const float* __restrict__ W  = p.W[lay];
            const float* __restrict__ Bv = p.B[lay];
            dst[t] = fast_tanh(fmaf(W[2 * t], src[2 * t],
                               fmaf(W[2 * t + 1], src[2 * t + 1], Bv[t])));
        }
        __syncthreads();
        float* tmp = src; src = dst; dst = tmp;
    }
    if (t == 0) out[row] = src[0];
}

extern "C" void kernel_launch(void* const* d_in, const int* in_sizes, int n_in,
                              void* d_out, int out_size, void* d_ws, size_t ws_size,
                              hipStream_t stream) {
    (void)d_ws; (void)ws_size; (void)n_in; (void)out_size;
    const float* x = (const float*)d_in[0];
    TreeParams p;
    for (int k = 0; k < TDEPTH; ++k) {
        p.W[k] = (const float*)d_in[1 + k];            // W_layers[0..13]
        p.B[k] = (const float*)d_in[1 + TDEPTH + k];   // b_layers[0..13]
    }
    const int rows = in_sizes[0] / NLEAVES;            // 4096
    btt_tree_kernel<<<rows, TBLOCK, 0, stream>>>(x, p, (float*)d_out);
}